// FOLinear_360777252983
// MI455X (gfx1250) — compile-verified
//
#include <hip/hip_runtime.h>

// ---------------------------------------------------------------------------
// Fused FOLinear for gfx1250 (MI455X), v3.
//  - Butterflies folded into fixed 64x64 complex matrices F and R (kernel A),
//    stored PAIR-INTERLEAVED so every WMMA B-fragment is one b64 load.
//  - S pre-swizzled into fragment-paired planes (kernel A).
//  - Kernel B: per 16-row tile: z = x*F^T, U_k = Z_k*S_k^T, y = ur*Rr^T-ui*Ri^T,
//    all V_WMMA_F32_16X16X4_F32 (640 static wmma), bias folded into rev store.
//  - x->LDS and LDS->y staging via CDNA5 async global<->LDS ops (ASYNCcnt).
// HBM traffic 64 MB -> ~2.75us floor @ 23.3 TB/s; FLOPs ride the WMMA pipe.
// ---------------------------------------------------------------------------

#define NTHREADS 256
#define TILE_B   16
#define ZROW     2052   // padded float stride per batch row in LDS
#define NLV      6

// d_ws float offsets (192KB total)
// Matrices stored paired: M_T[((i>>1)*64 + j)*2 + (i&1)] = M[j][i]
#define WS_FRT   0
#define WS_FIT   4096
#define WS_RRT   8192
#define WS_RMIT  12288     // -Ri
// S planes paired over q: SP[( (k*8 + qh)*16 + p )*2 + {0,1}] = S[p][2qh+{0,1}][k][c]
#define WS_SPR   16384
#define WS_SPI   32768
#define WS_FLOATS 49152    // 192 KB

typedef float v2f __attribute__((ext_vector_type(2)));
typedef float v8f __attribute__((ext_vector_type(8)));

__device__ __forceinline__ v8f wmma_f32_16x16x4(v2f a, v2f b, v8f c) {
    return __builtin_amdgcn_wmma_f32_16x16x4_f32(false, a, false, b, (short)0, c, false, false);
}

// ---- compile-time index helpers ----
__host__ __device__ constexpr int bitrev6(int i) {
    int r = 0;
    for (int b = 0; b < 6; ++b) r |= ((i >> b) & 1) << (5 - b);
    return r;
}
__host__ __device__ constexpr int fwd_perm(int l, int i) {
    int bs = 1 << (l + 2); int h = bs >> 1;
    int base = i & ~(bs - 1); int r = i & (bs - 1);
    return base + (r & 1) * h + (r >> 1);
}
__host__ __device__ constexpr int inv_perm(int l, int i) {
    int bs = 1 << (l + 2); int h = bs >> 1;
    int base = i & ~(bs - 1); int r = i & (bs - 1);
    return base + 2 * (r & (h - 1)) + (r / h);
}

__device__ __forceinline__ void apply_level(float (&xr)[64], float (&xi)[64], const float* w) {
#pragma unroll
    for (int p = 0; p < 32; ++p) {
        float4 wr = *(const float4*)(w + p * 8);      // {c1, -s2, -s1, c2}
        float4 wi = *(const float4*)(w + p * 8 + 4);  // {s1,  c2,  c1, s2}
        float xr0 = xr[2*p], xr1 = xr[2*p+1];
        float xi0 = xi[2*p], xi1 = xi[2*p+1];
        xr[2*p]   = wr.x*xr0 + wr.y*xr1 - (wi.x*xi0 + wi.y*xi1);
        xr[2*p+1] = wr.z*xr0 + wr.w*xr1 - (wi.z*xi0 + wi.w*xi1);
        xi[2*p]   = wr.x*xi0 + wr.y*xi1 + (wi.x*xr0 + wi.y*xr1);
        xi[2*p+1] = wr.z*xi0 + wr.w*xi1 + (wi.z*xr0 + wi.w*xr1);
    }
}

template <bool FWD>
__device__ __forceinline__ void permute64(float (&xr)[64], float (&xi)[64], int l) {
    float tr[64], ti[64];
#pragma unroll
    for (int i = 0; i < 64; ++i) {
        const int s = FWD ? fwd_perm(l, i) : inv_perm(l, i);
        tr[i] = xr[s]; ti[i] = xi[s];
    }
#pragma unroll
    for (int i = 0; i < 64; ++i) { xr[i] = tr[i]; xi[i] = ti[i]; }
}

__device__ __forceinline__ void phase_shift64(float (&xr)[64], float (&xi)[64], const float* ps) {
#pragma unroll
    for (int k = 0; k < 64; ++k) {
        float2 cs = *(const float2*)(ps + 2 * k);
        float r = cs.x * xr[k] - cs.y * xi[k];
        float i = cs.x * xi[k] + cs.y * xr[k];
        xr[k] = r; xi[k] = i;
    }
}

// ===========================================================================
// Kernel A: build F / R matrices (pair-interleaved) + pre-swizzle S planes.
// 1 block, 128 threads.
// ===========================================================================
__global__ __launch_bounds__(128, 1)
void build_tables(const float* __restrict__ ph_t,
                  const float* __restrict__ ph_tr,
                  const float* __restrict__ Sg,
                  float* __restrict__ ws)
{
    __shared__ float wf[1536], wv[1536], psf[128], psr[128];
    const int t = threadIdx.x;

    for (int i = t; i < 192; i += 128) {
        const int l = i >> 5, p = i & 31;
        {
            float p1 = ph_t[l*64 + p*2 + 0], p2 = ph_t[l*64 + p*2 + 1];
            float s1, c1, s2, c2;
            __sincosf(p1, &s1, &c1); __sincosf(p2, &s2, &c2);
            float* w = wf + i * 8;
            w[0]=c1; w[1]=-s2; w[2]=-s1; w[3]=c2; w[4]=s1; w[5]=c2; w[6]=c1; w[7]=s2;
        }
        {
            float p1 = ph_tr[l*64 + p*2 + 0], p2 = ph_tr[l*64 + p*2 + 1];
            float s1, c1, s2, c2;
            __sincosf(p1, &s1, &c1); __sincosf(p2, &s2, &c2);
            float* w = wv + i * 8;
            w[0]=c1; w[1]=-s2; w[2]=-s1; w[3]=c2; w[4]=s1; w[5]=c2; w[6]=c1; w[7]=s2;
        }
    }
    for (int i = t; i < 64; i += 128) {
        float s, c;
        __sincosf(ph_t [6*64 + i], &s, &c); psf[2*i] = c; psf[2*i+1] = s;
        __sincosf(ph_tr[6*64 + i], &s, &c); psr[2*i] = c; psr[2*i+1] = s;
    }

    // S fragment planes (no barrier needed; independent of wf/wv)
    for (int idx = t; idx < 8192; idx += 128) {
        const int k  = idx >> 7;
        const int qh = (idx >> 4) & 7;
        const int p  = idx & 15;
        const int q0 = 2 * qh, q1 = 2 * qh + 1;
        ws[WS_SPR + idx*2 + 0] = Sg[((p*16 + q0)*64 + k)*2 + 0];
        ws[WS_SPR + idx*2 + 1] = Sg[((p*16 + q1)*64 + k)*2 + 0];
        ws[WS_SPI + idx*2 + 0] = Sg[((p*16 + q0)*64 + k)*2 + 1];
        ws[WS_SPI + idx*2 + 1] = Sg[((p*16 + q1)*64 + k)*2 + 1];
    }
    __syncthreads();

    float xr[64], xi[64];
    if (t < 64) {
        const int c = t;   // column c of F
#pragma unroll
        for (int j = 0; j < 64; ++j) { xr[j] = (bitrev6(j) == c) ? 1.0f : 0.0f; xi[j] = 0.0f; }
#pragma unroll
        for (int l = 0; l < NLV; ++l) {
            apply_level(xr, xi, wf + l * 256);
            if (l < NLV - 1) permute64<true>(xr, xi, l);
        }
        phase_shift64(xr, xi, psf);
#pragma unroll
        for (int j = 0; j < 64; ++j) {   // pair-interleaved transpose store
            ws[WS_FRT + ((c >> 1)*64 + j)*2 + (c & 1)] = xr[j];
            ws[WS_FIT + ((c >> 1)*64 + j)*2 + (c & 1)] = xi[j];
        }
    } else {
        const int c = t - 64;  // column c of R
#pragma unroll
        for (int j = 0; j < 64; ++j) { xr[j] = (j == c) ? 1.0f : 0.0f; xi[j] = 0.0f; }
        phase_shift64(xr, xi, psr);
#pragma unroll
        for (int li = 0; li < NLV; ++li) {
            const int l = NLV - 1 - li;
            apply_level(xr, xi, wv + l * 256);
            if (l > 0) permute64<false>(xr, xi, l - 1);
        }
#pragma unroll
        for (int j = 0; j < 64; ++j) {   // output bit-reversal folded in
            ws[WS_RRT  + ((c >> 1)*64 + j)*2 + (c & 1)] =  xr[bitrev6(j)];
            ws[WS_RMIT + ((c >> 1)*64 + j)*2 + (c & 1)] = -xi[bitrev6(j)];
        }
    }
}

// ===========================================================================
// Kernel B: fused pipeline, one block = 16 batch rows, 8 waves.
// ===========================================================================
__global__ __launch_bounds__(NTHREADS, 1)
void folinear_main(const float* __restrict__ xin,
                   const float* __restrict__ bias,
                   const float* __restrict__ Wm,
                   float* __restrict__ yout)
{
    extern __shared__ float zbuf[];                   // 16 * ZROW floats
    const int t    = threadIdx.x;
    const int tile = blockIdx.x;
    const int lane = t & 31;
    const int wave = t >> 5;
    const int rN   = lane & 15;    // A: M(batch row) | B/D: N(col)
    const int hi   = lane >> 4;    // K-pair select {0,1} vs {2,3}

    const uint32_t lds0 = (uint32_t)(uintptr_t)zbuf;  // LDS byte offset (addr[31:0])

    // ---- stage 1: async global -> LDS staging of the x tile ----
    {
        const float* xt = xin + (size_t)tile * TILE_B * 1024;
#pragma unroll
        for (int i = 0; i < TILE_B; ++i) {
            uint32_t ldsoff = lds0 + 4u * (uint32_t)(i * ZROW + t * 4);
            uint32_t goff   = (uint32_t)(i * 4096 + t * 16);
            asm volatile("global_load_async_to_lds_b128 %0, %1, %2"
                         :: "v"(ldsoff), "v"(goff), "s"(xt) : "memory");
        }
        asm volatile("s_wait_asynccnt 0x0" ::: "memory");
    }
    __syncthreads();

    // ---- stage 2: fwd GEMM  z_q(16x64 cplx) = x_q(16x64 real) * F^T ----
    {
        const float* FrT = Wm + WS_FRT;
        const float* FiT = Wm + WS_FIT;
        v2f ax[2][16];
#pragma unroll
        for (int g2 = 0; g2 < 2; ++g2) {
            const int q = wave * 2 + g2;
#pragma unroll
            for (int ks = 0; ks < 16; ++ks) {
                const int i0 = ks * 4 + 2 * hi;
                float2 a = *(const float2*)(zbuf + rN * ZROW + q * 64 + i0);
                ax[g2][ks][0] = a.x; ax[g2][ks][1] = a.y;
            }
        }
        __syncthreads();   // z writes clobber x staging read by other waves

#pragma unroll
        for (int g2 = 0; g2 < 2; ++g2) {
            const int q = wave * 2 + g2;
#pragma unroll
            for (int nt = 0; nt < 4; ++nt) {
                v8f zr = {0,0,0,0,0,0,0,0};
                v8f zi = {0,0,0,0,0,0,0,0};
                const int j = nt * 16 + rN;
#pragma unroll
                for (int ks = 0; ks < 16; ++ks) {
                    const int pr = (ks * 2 + hi) * 64 + j;   // paired row
                    float2 fr = *(const float2*)(FrT + pr * 2);
                    float2 fi = *(const float2*)(FiT + pr * 2);
                    v2f br; br[0] = fr.x; br[1] = fr.y;
                    v2f bi; bi[0] = fi.x; bi[1] = fi.y;
                    zr = wmma_f32_16x16x4(ax[g2][ks], br, zr);
                    zi = wmma_f32_16x16x4(ax[g2][ks], bi, zi);
                }
#pragma unroll
                for (int j8 = 0; j8 < 8; ++j8) {
                    const int bo  = j8 + 8 * hi;
                    const int col = q * 64 + nt * 16 + rN;
                    *(float2*)(zbuf + bo * ZROW + col * 2) = make_float2(zr[j8], zi[j8]);
                }
            }
        }
    }
    __syncthreads();

    // ---- stage 3: mix  U_k = Z_k(16b x 16q) * S_k^T, complex, per k ----
    {
        const float* SPr = Wm + WS_SPR;
        const float* SPi = Wm + WS_SPI;
#pragma unroll
        for (int kk = 0; kk < 8; ++kk) {
            const int k = wave * 8 + kk;
            v8f ur = {0,0,0,0,0,0,0,0};
            v8f ui = {0,0,0,0,0,0,0,0};
#pragma unroll
            for (int qc = 0; qc < 16; qc += 4) {
                const int q0 = qc + 2 * hi;
                const int qh = (qc >> 1) + hi;
                float2 a0 = *(const float2*)(zbuf + rN * ZROW + ((q0    ) * 64 + k) * 2);
                float2 a1 = *(const float2*)(zbuf + rN * ZROW + ((q0 + 1) * 64 + k) * 2);
                v2f azr; azr[0] = a0.x; azr[1] = a1.x;
                v2f azi; azi[0] = a0.y; azi[1] = a1.y;
                const int sidx = ((k * 8 + qh) * 16 + rN) * 2;
                float2 sr = *(const float2*)(SPr + sidx);
                float2 si = *(const float2*)(SPi + sidx);
                v2f bsr;  bsr[0]  =  sr.x; bsr[1]  =  sr.y;
                v2f bsi;  bsi[0]  =  si.x; bsi[1]  =  si.y;
                v2f bnsi; bnsi[0] = -si.x; bnsi[1] = -si.y;
                ur = wmma_f32_16x16x4(azr, bsr,  ur);
                ur = wmma_f32_16x16x4(azi, bnsi, ur);
                ui = wmma_f32_16x16x4(azr, bsi,  ui);
                ui = wmma_f32_16x16x4(azi, bsr,  ui);
            }
#pragma unroll
            for (int j8 = 0; j8 < 8; ++j8) {
                const int bo = j8 + 8 * hi;
                *(float2*)(zbuf + bo * ZROW + (rN * 64 + k) * 2) = make_float2(ur[j8], ui[j8]);
            }
        }
    }
    __syncthreads();

    // ---- stage 4: rev GEMM  y_p(16x64 real) = ur*Rr^T - ui*Ri^T (+bias) ----
    {
        const float* RrT  = Wm + WS_RRT;
        const float* RmiT = Wm + WS_RMIT;
        float au[2][16][4];
#pragma unroll
        for (int g2 = 0; g2 < 2; ++g2) {
            const int p = wave * 2 + g2;
#pragma unroll
            for (int ks = 0; ks < 16; ++ks) {
                const int i0 = ks * 4 + 2 * hi;
                float2 u0 = *(const float2*)(zbuf + rN * ZROW + (p * 64 + i0    ) * 2);
                float2 u1 = *(const float2*)(zbuf + rN * ZROW + (p * 64 + i0 + 1) * 2);
                au[g2][ks][0] = u0.x; au[g2][ks][1] = u0.y;
                au[g2][ks][2] = u1.x; au[g2][ks][3] = u1.y;
            }
        }
        __syncthreads();   // y staging clobbers u read by other waves

#pragma unroll
        for (int g2 = 0; g2 < 2; ++g2) {
            const int p = wave * 2 + g2;
#pragma unroll
            for (int nt = 0; nt < 4; ++nt) {
                v8f y = {0,0,0,0,0,0,0,0};
                const int j   = nt * 16 + rN;
                const int col = p * 64 + j;
                const float bb = bias[col];
#pragma unroll
                for (int ks = 0; ks < 16; ++ks) {
                    const int pr = (ks * 2 + hi) * 64 + j;
                    float2 rr = *(const float2*)(RrT  + pr * 2);
                    float2 rm = *(const float2*)(RmiT + pr * 2);
                    v2f ar; ar[0] = au[g2][ks][0]; ar[1] = au[g2][ks][2];
                    v2f ai; ai[0] = au[g2][ks][1]; ai[1] = au[g2][ks][3];
                    v2f br; br[0] = rr.x; br[1] = rr.y;
                    v2f bm; bm[0] = rm.x; bm[1] = rm.y;
                    y = wmma_f32_16x16x4(ar, br, y);
                    y = wmma_f32_16x16x4(ai, bm, y);
                }
#pragma unroll
                for (int j8 = 0; j8 < 8; ++j8) {
                    const int bo = j8 + 8 * hi;
                    zbuf[bo * ZROW + col] = y[j8] + bb;
                }
            }
        }
    }
    __syncthreads();

    // ---- stage 5: async LDS -> global store ----
    {
        float* yt = yout + (size_t)tile * TILE_B * 1024;
#pragma unroll
        for (int i = 0; i < TILE_B; ++i) {
            uint32_t ldsoff = lds0 + 4u * (uint32_t)(i * ZROW + t * 4);
            uint32_t goff   = (uint32_t)(i * 4096 + t * 16);
            asm volatile("global_store_async_from_lds_b128 %0, %1, %2"
                         :: "v"(goff), "v"(ldsoff), "s"(yt) : "memory");
        }
        asm volatile("s_wait_asynccnt 0x0" ::: "memory");
    }
}

extern "C" void kernel_launch(void* const* d_in, const int* in_sizes, int n_in,
                              void* d_out, int out_size, void* d_ws, size_t ws_size,
                              hipStream_t stream) {
    const float* x     = (const float*)d_in[0];
    const float* ph_t  = (const float*)d_in[1];
    const float* ph_tr = (const float*)d_in[2];
    const float* S     = (const float*)d_in[3];
    const float* bias  = (const float*)d_in[4];
    float* out = (float*)d_out;
    float* ws  = (float*)d_ws;   // needs WS_FLOATS*4 = 192 KB

    const int Btot   = in_sizes[0] / 1024;
    const int nTiles = Btot / TILE_B;

    build_tables<<<1, 128, 0, stream>>>(ph_t, ph_tr, S, ws);

    const size_t shmem = (size_t)(TILE_B * ZROW) * sizeof(float);  // ~128 KB
    (void)hipFuncSetAttribute((const void*)folinear_main,
                              hipFuncAttributeMaxDynamicSharedMemorySize,
                              (int)shmem);
    folinear_main<<<nTiles, NTHREADS, shmem, stream>>>(x, bias, ws, out);
}